// EnhancedBCMLayer_69148973466158
// MI455X (gfx1250) — compile-verified
//
#include <hip/hip_runtime.h>

// out = x (4096x2048) * W^T + b, W block-circulant from index_vectors:
//   W[f*16+n, g*16+m] = p[f, g, (n-m) & 15]
// Tiled fp32 WMMA GEMM (v_wmma_f32_16x16x4_f32); circulant B-tiles generated
// on the fly from 16-float generators in LDS; x staged via gfx1250 async
// global->LDS loads (ASYNCcnt path).

#define IN_F   2048
#define OUT_F  2048
#define BATCH  4096
#define GB     128      // input (g) blocks of 16
#define FB     128      // output (f) blocks of 16
#define XPITCH 132      // 128 cols + 4 pad floats: kills LDS bank conflicts on transposed reads

typedef __attribute__((ext_vector_type(2))) float v2f;
typedef __attribute__((ext_vector_type(8))) float v8f;
typedef __attribute__((ext_vector_type(4))) int   v4i;

#define AS1 __attribute__((address_space(1)))
#define AS3 __attribute__((address_space(3)))

#if defined(__AMDGCN__) && __has_builtin(__builtin_amdgcn_global_load_async_to_lds_b128)
#define HAVE_ASYNC_LDS 1
#endif

__device__ __forceinline__ void wait_asynccnt0() {
#if defined(HAVE_ASYNC_LDS)
#if __has_builtin(__builtin_amdgcn_s_wait_asynccnt)
    __builtin_amdgcn_s_wait_asynccnt(0);
#else
    asm volatile("s_wait_asynccnt 0x0" ::: "memory");
#endif
#endif
}

__global__ __launch_bounds__(128) void bcm_wmma_f32_kernel(
    const float* __restrict__ x,      // [BATCH][IN_F]
    const float* __restrict__ p,      // [FB][GB][16]
    const float* __restrict__ bias,   // [OUT_F]
    float* __restrict__ out)          // [BATCH][OUT_F]
{
    __shared__ float p_lds[4 * GB * 16];   // 32 KB: generators for this block's 4 f-tiles
    __shared__ float x_lds[64 * XPITCH];   // ~33 KB: one 64x128 chunk of x

    const int tid  = threadIdx.x;
    const int wave = tid >> 5;             // 0..3 (wave32)
    const int lane = tid & 31;
    const int ln   = lane & 15;            // N / M-sub index
    const int lh   = lane >> 4;            // lane half

    const int n0 = blockIdx.x * 64;        // out-feature base (32 blocks)
    const int m0 = blockIdx.y * 64;        // batch base      (64 blocks)
    const int f0 = blockIdx.x * 4;         // first f-block of this workgroup

    // ---- stage the 4 generator slabs p[f0+fl][*][*] (16B-aligned contiguous copies) ----
#if defined(HAVE_ASYNC_LDS)
    for (int i = tid * 4; i < 4 * GB * 16; i += 128 * 4) {
        const int fl  = i >> 11;           // /2048
        const int rem = i & 2047;
        __builtin_amdgcn_global_load_async_to_lds_b128(
            (AS1 v4i*)(p + (size_t)(f0 + fl) * (GB * 16) + rem),
            (AS3 v4i*)&p_lds[i], 0, 0);
    }
#else
    for (int i = tid; i < 4 * GB * 16; i += 128) {
        const int fl  = i >> 11;
        const int rem = i & 2047;
        p_lds[i] = p[(size_t)(f0 + fl) * (GB * 16) + rem];
    }
#endif

    v8f acc[4] = {};                       // 4 output tiles (16x64 strip per wave)

    const int c4 = (tid & 31) * 4;         // column group for cooperative x staging
    const int r0 = tid >> 5;               // row offset for cooperative x staging

    for (int kc = 0; kc < 16; ++kc) {      // 16 chunks of 128 input columns
        __syncthreads();                   // previous chunk fully consumed (also fences p_lds on kc==0)

        // ---- stage 64x128 x chunk (coalesced 16B per lane) ----
        #pragma unroll
        for (int r = 0; r < 64; r += 4) {
#if defined(HAVE_ASYNC_LDS)
            __builtin_amdgcn_global_load_async_to_lds_b128(
                (AS1 v4i*)(x + (size_t)(m0 + r + r0) * IN_F + kc * 128 + c4),
                (AS3 v4i*)&x_lds[(r + r0) * XPITCH + c4], 0, 0);
#else
            const float4 v = *(const float4*)(x + (size_t)(m0 + r + r0) * IN_F + kc * 128 + c4);
            *(float4*)&x_lds[(r + r0) * XPITCH + c4] = v;
#endif
        }
        if (kc + 1 < 16) {                 // uniform branch; gfx1250 global_prefetch path
            __builtin_prefetch(x + (size_t)(m0 + r0) * IN_F + (kc + 1) * 128 + c4, 0, 1);
        }
        wait_asynccnt0();                  // own async transfers (incl. p slab on kc==0) landed
        __syncthreads();                   // everyone's transfers visible

        for (int gl = 0; gl < 8; ++gl) {   // 8 g-blocks per chunk
            const int g = kc * 8 + gl;
            const float* prow0 = &p_lds[(0 * GB + g) * 16];
            const float* prow1 = &p_lds[(1 * GB + g) * 16];
            const float* prow2 = &p_lds[(2 * GB + g) * 16];
            const float* prow3 = &p_lds[(3 * GB + g) * 16];

            #pragma unroll
            for (int q = 0; q < 4; ++q) {  // K=16 per g-block -> 4 x (K=4) WMMAs
                // A fragment (16x4 f32): lane ln = batch row, k = q*4 + 2*lh + {0,1}
                const int colL = gl * 16 + q * 4 + 2 * lh;
                const v2f a = *(const v2f*)&x_lds[(wave * 16 + ln) * XPITCH + colL];

                // B fragment (4x16 f32): B[k][n] = p[f,g,(n-m)&15], m = q*4+2*lh+{0,1}, n = ln
                const int m  = q * 4 + 2 * lh;
                const int i0 = (ln - m) & 15;
                const int i1 = (ln - m - 1) & 15;
                v2f b0; b0.x = prow0[i0]; b0.y = prow0[i1];
                v2f b1; b1.x = prow1[i0]; b1.y = prow1[i1];
                v2f b2; b2.x = prow2[i0]; b2.y = prow2[i1];
                v2f b3; b3.x = prow3[i0]; b3.y = prow3[i1];

                acc[0] = __builtin_amdgcn_wmma_f32_16x16x4_f32(false, a, false, b0, (short)0, acc[0], false, false);
                acc[1] = __builtin_amdgcn_wmma_f32_16x16x4_f32(false, a, false, b1, (short)0, acc[1], false, false);
                acc[2] = __builtin_amdgcn_wmma_f32_16x16x4_f32(false, a, false, b2, (short)0, acc[2], false, false);
                acc[3] = __builtin_amdgcn_wmma_f32_16x16x4_f32(false, a, false, b3, (short)0, acc[3], false, false);
            }
        }
    }

    // ---- epilogue: C/D layout M = r + 8*lh, N = ln; add bias, store ----
    const int rowBase = m0 + wave * 16 + lh * 8;
    #pragma unroll
    for (int fl = 0; fl < 4; ++fl) {
        const int n  = n0 + fl * 16 + ln;
        const float bv = bias[n];
        float* po = out + (size_t)rowBase * OUT_F + n;
        #pragma unroll
        for (int r = 0; r < 8; ++r)
            po[(size_t)r * OUT_F] = acc[fl][r] + bv;
    }
}

extern "C" void kernel_launch(void* const* d_in, const int* in_sizes, int n_in,
                              void* d_out, int out_size, void* d_ws, size_t ws_size,
                              hipStream_t stream) {
    (void)in_sizes; (void)n_in; (void)d_ws; (void)ws_size; (void)out_size;
    const float* x    = (const float*)d_in[0];   // 4096*2048 f32
    const float* p    = (const float*)d_in[1];   // 128*128*16 f32
    const float* bias = (const float*)d_in[2];   // 2048 f32
    float* out        = (float*)d_out;           // 4096*2048 f32

    dim3 grid(OUT_F / 64, BATCH / 64);           // (32, 64)
    dim3 block(128);                             // 4 waves (wave32)
    bcm_wmma_f32_kernel<<<grid, block, 0, stream>>>(x, p, bias, out);
}